// ProjectQaudMesh2Image_61795989455180
// MI455X (gfx1250) — compile-verified
//
#include <hip/hip_runtime.h>

#define HH 192
#define WW 192
#define NPIX (HH*WW)
#define FARV 10.0f
#define BSZ 4
#define NSQ 12
#define CFN 16
#define QSTRIDE 48

// d_out flat layout (reference return order: image, alpha, depth, area, center_depth)
#define IMG_OFF   0
#define ALPHA_OFF (BSZ*CFN*NPIX)           // 2359296
#define DEPTH_OFF (ALPHA_OFF + BSZ*NPIX)   // 2506752
#define AREA_OFF  (DEPTH_OFF + BSZ*NPIX)   // 2654208
#define CD_OFF    (AREA_OFF + BSZ)         // 2654212

typedef __attribute__((ext_vector_type(2))) float v2f;
typedef __attribute__((ext_vector_type(8))) float v8f;

// ---------------------------------------------------------------------------
// Kernel 1: per-quad geometry constants + area / center_depth reductions.
// One thread per quad (48 quads total).
// Per-quad float[48] layout in workspace:
//  [0..2] nrm  [3] p0·nrm  [4..15] corners  [16..27] edges  [28..31] 1/(|e|+1e-3)
//  [32..40] Rm (row-major)  [41..43] Tm  [44] 1/sx  [45] 1/sy
// ---------------------------------------------------------------------------
__global__ __launch_bounds__(64) void quad_setup_kernel(
    const float* __restrict__ K, const float* __restrict__ RT,
    const float* __restrict__ quad_RT, const float* __restrict__ quad_scale,
    float* __restrict__ qc, float* __restrict__ out)
{
  __shared__ float sArea[64];
  __shared__ float sCd[64];
  const int m = threadIdx.x;
  float areaPart = 0.0f, cdPart = 0.0f;
  if (m < BSZ*NSQ) {
    const int b = m / NSQ, n = m % NSQ;
    float rt[16];
    #pragma unroll
    for (int i = 0; i < 12; ++i) rt[i] = RT[b*12 + i];
    rt[12] = 0.0f; rt[13] = 0.0f; rt[14] = 0.0f; rt[15] = 1.0f;
    float qr[16];
    #pragma unroll
    for (int i = 0; i < 16; ++i) qr[i] = quad_RT[(b*NSQ + n)*16 + i];
    float q2c[16];
    #pragma unroll
    for (int i = 0; i < 4; ++i)
      #pragma unroll
      for (int j = 0; j < 4; ++j) {
        float s = 0.0f;
        #pragma unroll
        for (int k = 0; k < 4; ++k) s += rt[i*4+k]*qr[k*4+j];
        q2c[i*4+j] = s;
      }
    const float sx = quad_scale[(b*NSQ + n)*2 + 0];
    const float sy = quad_scale[(b*NSQ + n)*2 + 1];
    const float ch[4][4] = {{ sx,0.f, sy,1.f},{ sx,0.f,-sy,1.f},
                            {-sx,0.f,-sy,1.f},{-sx,0.f, sy,1.f}};
    float c[4][3];
    #pragma unroll
    for (int k = 0; k < 4; ++k)
      #pragma unroll
      for (int i = 0; i < 3; ++i)
        c[k][i] = q2c[i*4+0]*ch[k][0] + q2c[i*4+1]*ch[k][1]
                + q2c[i*4+2]*ch[k][2] + q2c[i*4+3]*ch[k][3];
    float e10[3], e30[3], nr[3];
    #pragma unroll
    for (int i = 0; i < 3; ++i) { e10[i] = c[1][i]-c[0][i]; e30[i] = c[3][i]-c[0][i]; }
    nr[0] = e10[1]*e30[2] - e10[2]*e30[1];
    nr[1] = e10[2]*e30[0] - e10[0]*e30[2];
    nr[2] = e10[0]*e30[1] - e10[1]*e30[0];
    const float nn = sqrtf(nr[0]*nr[0] + nr[1]*nr[1] + nr[2]*nr[2]) + 1e-6f;
    nr[0] /= nn; nr[1] /= nn; nr[2] /= nn;
    float p0[3];
    #pragma unroll
    for (int i = 0; i < 3; ++i) p0[i] = 0.25f*(c[0][i]+c[1][i]+c[2][i]+c[3][i]);
    const float pdotn = p0[0]*nr[0] + p0[1]*nr[1] + p0[2]*nr[2];
    float* q = qc + m*QSTRIDE;
    q[0] = nr[0]; q[1] = nr[1]; q[2] = nr[2]; q[3] = pdotn;
    #pragma unroll
    for (int k = 0; k < 4; ++k)
      #pragma unroll
      for (int i = 0; i < 3; ++i) q[4 + k*3 + i] = c[k][i];
    #pragma unroll
    for (int k = 0; k < 4; ++k) {
      const int k2 = (k+1) & 3;
      const float ex = c[k2][0]-c[k][0], ey = c[k2][1]-c[k][1], ez = c[k2][2]-c[k][2];
      q[16 + k*3 + 0] = ex; q[16 + k*3 + 1] = ey; q[16 + k*3 + 2] = ez;
      q[28 + k] = 1.0f / (sqrtf(ex*ex + ey*ey + ez*ez) + 0.001f);
    }
    #pragma unroll
    for (int i = 0; i < 3; ++i)
      #pragma unroll
      for (int j = 0; j < 3; ++j)
        q[32 + i*3 + j] = q2c[j*4 + i];          // Rm = q2c[:3,:3]^T
    #pragma unroll
    for (int i = 0; i < 3; ++i)
      q[41 + i] = -(q2c[0*4+i]*q2c[0*4+3] + q2c[1*4+i]*q2c[1*4+3] + q2c[2*4+i]*q2c[2*4+3]);
    q[44] = 1.0f / sx; q[45] = 1.0f / sy; q[46] = 0.0f; q[47] = 0.0f;
    // area / center depth partials
    float u2[4], v2[4];
    #pragma unroll
    for (int k = 0; k < 4; ++k) {
      const float px = K[0]*c[k][0] + K[1]*c[k][1] + K[2]*c[k][2];
      const float py = K[3]*c[k][0] + K[4]*c[k][1] + K[5]*c[k][2];
      const float pz = K[6]*c[k][0] + K[7]*c[k][1] + K[8]*c[k][2];
      const float inv = 1.0f / (pz + 1e-6f);
      u2[k] = px*inv; v2[k] = py*inv;
    }
    float asum = 0.0f;
    #pragma unroll
    for (int k = 0; k < 4; ++k) {
      const int j = (k+1) & 3;
      asum += fabsf(u2[k]*v2[j] - v2[k]*u2[j]);
    }
    areaPart = (asum * 0.25f) / (float)(WW*HH);
    cdPart   = 0.25f * (c[0][2] + c[1][2] + c[2][2] + c[3][2]);
  }
  sArea[threadIdx.x] = areaPart;
  sCd[threadIdx.x]   = cdPart;
  __syncthreads();
  if (threadIdx.x < BSZ) {
    float a = 0.0f, cd = 0.0f;
    #pragma unroll
    for (int n = 0; n < NSQ; ++n) { a += sArea[threadIdx.x*NSQ + n]; cd += sCd[threadIdx.x*NSQ + n]; }
    out[AREA_OFF + threadIdx.x] = a * 0.5f;
    out[CD_OFF + threadIdx.x]   = cd / (float)NSQ;
  }
}

// ---------------------------------------------------------------------------
// Kernel 2: one pixel per lane; WMMA f32 16x16x4 blend over texel space.
// grid = (NPIX/256, BS), block = 256 (8 wave32s).
// __launch_bounds__(256, 4): cap at 256 VGPRs/lane -> >=4 waves/SIMD32,
// no VGPR-MSB switching overhead.
// ---------------------------------------------------------------------------
__global__ __launch_bounds__(256, 4) void render_kernel(
    const float* __restrict__ K, const float* __restrict__ quad_feat,
    const float* __restrict__ qc, float* __restrict__ out)
{
  __shared__ __align__(16) float sQ[NSQ*QSTRIDE];   // 576 floats
  __shared__ __align__(16) float sW[8*52*32];       // per-wave 52x32 weight matrix

  const int tid  = threadIdx.x;
  const int b    = blockIdx.y;
  const int lane = tid & 31;
  const int wid  = tid >> 5;

  // --- async-stage the batch's quad constants into LDS (CDNA5 async path) ---
  {
    const unsigned lbase = (unsigned)(size_t)(&sQ[0]);
    const float* src = qc + b*NSQ*QSTRIDE;
    for (int i = tid; i < (NSQ*QSTRIDE)/4; i += 256) {      // 144 b128 chunks
      unsigned la = lbase + (unsigned)(i*16);
      unsigned va = (unsigned)(i*16);
      asm volatile("global_load_async_to_lds_b128 %0, %1, %2"
                   :: "v"(la), "v"(va), "s"(src) : "memory");
    }
    asm volatile("s_wait_asynccnt 0" ::: "memory");
  }
  __syncthreads();

  const int pixBlock = blockIdx.x * 256;
  const int pix = pixBlock + tid;
  const int px = pix % WW, py = pix / WW;
  const float fx = K[0], cxk = K[2], fy = K[4], cyk = K[5];
  const float lx = -(((float)px + 0.5f) - cxk) / fx;
  const float ly = -(((float)py + 0.5f) - cyk) / fy;
  const float lz = -1.0f;

  float uq[NSQ], vq[NSQ], wr[NSQ];
  unsigned mnear = 0u;
  float prodOneMinus = 1.0f;
  float dmin = FARV;
  float wsum = 0.0f;

  // ---------------- pass 1: per-quad geometry ----------------
  #pragma unroll
  for (int n = 0; n < NSQ; ++n) {
    const float* q = &sQ[n*QSTRIDE];
    const float nx = q[0], ny = q[1], nz = q[2], pdn = q[3];
    const float demon = lx*nx + ly*ny + lz*nz;
    const float t = pdn / (demon + 1e-6f);
    const float pxx = t*lx, pyy = t*ly, pzz = t*lz;
    const float mfront = (t < 0.0f) ? 1.0f : 0.0f;
    float dis = 1e30f;
    bool onb = true;
    #pragma unroll
    for (int ke = 0; ke < 4; ++ke) {
      const int k2 = (ke+1) & 3;
      const float c0x = q[4+ke*3], c0y = q[5+ke*3], c0z = q[6+ke*3];
      const float c1x = q[4+k2*3], c1y = q[5+k2*3], c1z = q[6+k2*3];
      const float ex = q[16+ke*3], ey = q[17+ke*3], ez = q[18+ke*3];
      const float dx0 = pxx-c0x, dy0 = pyy-c0y, dz0 = pzz-c0z;
      const float crx = ey*dz0 - ez*dy0;
      const float cry = ez*dx0 - ex*dz0;
      const float crz = ex*dy0 - ey*dx0;
      onb = onb && ((crx*nx + cry*ny + crz*nz) < 0.0f);
      const float d0 = dx0*ex + dy0*ey + dz0*ez;
      const float dx1 = pxx-c1x, dy1 = pyy-c1y, dz1 = pzz-c1z;
      const float d1 = -(dx1*ex + dy1*ey + dz1*ez);
      const float n0 = sqrtf(dx0*dx0 + dy0*dy0 + dz0*dz0);
      const float n1 = sqrtf(dx1*dx1 + dy1*dy1 + dz1*dz1);
      const float dend = (d0 < 0.0f ? n0 : 0.0f) + (d1 < 0.0f ? n1 : 0.0f);
      const float ncr = sqrtf(crx*crx + cry*cry + crz*crz);
      const float dseg = (d0 > 0.0f && d1 > 0.0f) ? ncr*q[28+ke] : dend;
      dis = fminf(dis, dseg);
    }
    const float dis2 = dis*dis;
    const float disT = onb ? (-FARV*dis2) : dis2;
    const float alpha = mfront / (1.0f + __expf(disT * 1e4f));
    prodOneMinus *= (1.0f - alpha);
    const float depthf = onb ? pzz : 1e6f;
    const float depthc = fminf(fmaxf(depthf, 0.0f), FARV);
    dmin = fminf(dmin, depthc);
    if (depthf > 0.0f) mnear |= (1u << n);
    const float invd = (FARV - depthc) * (1.0f/FARV);
    const float w = alpha * __expf(invd * 20.0f);       // 1/GAMMA = 20
    wr[n] = w; wsum += w;
    const float qx = q[32]*pxx + q[33]*pyy + q[34]*pzz + q[41];
    const float qz = q[38]*pxx + q[39]*pyy + q[40]*pzz + q[43];
    uq[n] = qx * q[44];
    vq[n] = qz * q[45];
  }
  out[ALPHA_OFF + b*NPIX + pix] = 1.0f - prodOneMinus;
  out[DEPTH_OFF + b*NPIX + pix] = dmin;

  // ---------------- pass 2: WMMA blend D[16ch x 16pix] += A[16x4] * B[4x16] ----------------
  const float invDen = 1.0f / (wsum + 0.1f);
  float* WL = &sW[wid * 52 * 32];
  v8f acc0 = {}; v8f acc1 = {};
  const int col   = lane & 15;   // A: channel / B,D: N column
  const int lhalf = lane >> 4;   // K-half selector per ISA VGPR layout
  const float* Fbase = quad_feat + (size_t)b*NSQ*49*CFN;

  #pragma unroll
  for (int n = 0; n < NSQ; ++n) {
    // zero this wave's 52x32 weight tile (rows 49..51 are zero-A pads)
    const float4 z = {0.0f, 0.0f, 0.0f, 0.0f};
    #pragma unroll
    for (int i = 0; i < 13; ++i)
      *((float4*)&WL[(lane + 32*i) * 4]) = z;

    // bilinear weights for this lane's pixel
    const float wgt = wr[n] * invDen * (((mnear >> n) & 1u) ? 1.0f : 0.0f);
    const float gx = (uq[n] + 1.0f)*3.5f - 0.5f;
    const float gy = (vq[n] + 1.0f)*3.5f - 0.5f;
    const float x0f = floorf(gx), y0f = floorf(gy);
    const float wx = gx - x0f, wy = gy - y0f;
    const int x0i = (int)fminf(fmaxf(x0f,        0.0f), 6.0f);
    const int x1i = (int)fminf(fmaxf(x0f + 1.0f, 0.0f), 6.0f);
    const int y0i = (int)fminf(fmaxf(y0f,        0.0f), 6.0f);
    const int y1i = (int)fminf(fmaxf(y0f + 1.0f, 0.0f), 6.0f);
    int t0 = y0i*7 + x0i, t1 = y0i*7 + x1i, t2 = y1i*7 + x0i, t3 = y1i*7 + x1i;
    float w0 = wgt*(1.0f-wy)*(1.0f-wx), w1 = wgt*(1.0f-wy)*wx;
    float w2 = wgt*wy*(1.0f-wx),        w3 = wgt*wy*wx;
    // merge border-clamp duplicates; redirect dups to pad rows (A is zero there)
    if (t1 == t0) { w0 += w1; t1 = 49; }
    if (t2 == t0) { w0 += w2; t2 = 50; } else if (t2 == t1) { w1 += w2; t2 = 50; }
    if (t3 == t0) { w0 += w3; t3 = 51; } else if (t3 == t1) { w1 += w3; t3 = 51; }
    else if (t3 == t2) { w2 += w3; t3 = 51; }
    WL[t0*32 + lane] = w0; WL[t1*32 + lane] = w1;
    WL[t2*32 + lane] = w2; WL[t3*32 + lane] = w3;

    const float* Fq = Fbase + n*49*CFN;
    #pragma unroll 2
    for (int kk = 0; kk < 13; ++kk) {
      const int kb = kk*4 + lhalf*2;
      v2f a;
      a.x = (kb     < 49) ? Fq[kb*CFN + col]     : 0.0f;
      a.y = (kb + 1 < 49) ? Fq[(kb+1)*CFN + col] : 0.0f;
      v2f b0, b1;
      b0.x = WL[kb*32 + col];        b0.y = WL[(kb+1)*32 + col];
      b1.x = WL[kb*32 + 16 + col];   b1.y = WL[(kb+1)*32 + 16 + col];
      acc0 = __builtin_amdgcn_wmma_f32_16x16x4_f32(false, a, false, b0, (short)0, acc0, false, false);
      acc1 = __builtin_amdgcn_wmma_f32_16x16x4_f32(false, a, false, b1, (short)0, acc1, false, false);
    }
  }

  // D layout: VGPR j -> channel = 8*lhalf + j, pixel = N = col (per tile)
  const int pixbase = pixBlock + wid*32;
  #pragma unroll
  for (int j = 0; j < 8; ++j) {
    const int c = lhalf*8 + j;
    const size_t o = (size_t)(b*CFN + c)*NPIX + pixbase;
    out[IMG_OFF + o + col]      = acc0[j];
    out[IMG_OFF + o + 16 + col] = acc1[j];
  }
}

extern "C" void kernel_launch(void* const* d_in, const int* in_sizes, int n_in,
                              void* d_out, int out_size, void* d_ws, size_t ws_size,
                              hipStream_t stream) {
  const float* K          = (const float*)d_in[0];
  const float* RT         = (const float*)d_in[1];
  const float* quad_RT    = (const float*)d_in[2];
  const float* quad_feat  = (const float*)d_in[3];
  const float* quad_scale = (const float*)d_in[4];
  float* out = (float*)d_out;
  float* qc  = (float*)d_ws;   // 48 quads x 48 floats = 9216 B

  quad_setup_kernel<<<1, 64, 0, stream>>>(K, RT, quad_RT, quad_scale, qc, out);
  render_kernel<<<dim3(NPIX/256, BSZ), 256, 0, stream>>>(K, quad_feat, qc, out);
}